// Spiking1DeLIFLayer_86019605004461
// MI455X (gfx1250) — compile-verified
//
#include <hip/hip_runtime.h>

// Spiking LIF layer, (B,C,T)=(64,512,1024), sequential scan over T.
// Memory-bound: ~268 MB traffic -> ~11.5us floor at 23.3 TB/s.
// Strategy: 1 wave per 32 rows; stream time-chunks via CDNA5 async
// global->LDS copies (double buffered, s_wait_asynccnt), compute the
// recurrence from LDS (padded rows, 2-way max bank conflict), stage
// spikes in LDS and flush with coalesced b128 stores.

typedef __attribute__((ext_vector_type(4))) float v4f;

#define B_     64
#define C_     512
#define T_     1024
#define TC     64            // time-steps per chunk (256 B per row per chunk)
#define STR    68            // LDS row stride in floats: 272 B, 16B-aligned, gcd(68,64)=4
#define NROW   32            // rows per block == wave32
#define NCHUNK (T_ / TC)     // 16

// CDNA5 async copy: global -> LDS, 16 bytes per lane, tracked by ASYNCcnt.
__device__ __forceinline__ void async_ld_b128(unsigned lds_byte_off, const float* gaddr) {
    asm volatile("global_load_async_to_lds_b128 %0, %1, off"
                 :: "v"(lds_byte_off), "v"(gaddr)
                 : "memory");
}
__device__ __forceinline__ void wait_async_le16() {
    asm volatile("s_wait_asynccnt 16" ::: "memory");
}
__device__ __forceinline__ void wait_async_0() {
    asm volatile("s_wait_asynccnt 0" ::: "memory");
}
__device__ __forceinline__ void cfence() { asm volatile("" ::: "memory"); }

__global__ __launch_bounds__(NROW)
void lif_scan_kernel(const float* __restrict__ x,
                     const float* __restrict__ beta_p,
                     const float* __restrict__ vth,
                     float* __restrict__ out) {
    __shared__ float s_in[2][NROW * STR];   // 2 x 8704 B, async-load destination
    __shared__ float s_out[NROW * STR];     // 8704 B, spike staging

    const int lane  = threadIdx.x;          // 0..31 (wave32)
    const int row0  = blockIdx.x * NROW;    // first (b,c) row of this block
    const int row   = row0 + lane;          // this lane's private row

    const float beta = beta_p[0];           // uniform -> scalar load
    const float vthc = vth[row & (C_ - 1)]; // c = row % C

    // LDS byte offset of the shared buffers (low 32 bits of the generic
    // pointer to an LDS object are the workgroup-relative LDS address).
    const unsigned in_base  = (unsigned)(unsigned long long)&s_in[0][0];

    // Cooperative tile mapping: per j, lanes 0-15 cover row 2j (256 B
    // contiguous), lanes 16-31 cover row 2j+1 (256 B contiguous).
    const int vlane = lane & 15;            // float4 slot within a row chunk
    const int rhalf = lane >> 4;            // 0 or 1

    const float* xblock = x   + (size_t)row0 * T_;
    float*       oblock = out + (size_t)row0 * T_;

    float mem = 0.0f;
    float spk = 0.0f;

    // ---- prefetch chunk 0 into buffer 0 (16 async b128 per wave) ----
    {
        const float* gb = xblock;
#pragma unroll
        for (int j = 0; j < 16; ++j) {
            const int r = 2 * j + rhalf;
            async_ld_b128(in_base + (unsigned)((r * STR + vlane * 4) * 4),
                          gb + (size_t)r * T_ + vlane * 4);
        }
    }

#pragma unroll 1
    for (int k = 0; k < NCHUNK; ++k) {
        const int p = k & 1;

        // ---- prefetch chunk k+1 into the other buffer, then wait for chunk k ----
        if (k + 1 < NCHUNK) {
            const float*   gb  = xblock + (k + 1) * TC;
            const unsigned dst = in_base + (unsigned)((1 - p) * NROW * STR * 4);
#pragma unroll
            for (int j = 0; j < 16; ++j) {
                const int r = 2 * j + rhalf;
                async_ld_b128(dst + (unsigned)((r * STR + vlane * 4) * 4),
                              gb + (size_t)r * T_ + vlane * 4);
            }
            wait_async_le16();   // loads complete in order: chunk k is in LDS
        } else {
            wait_async_0();
        }

        // ---- compute: 64 sequential LIF steps on this lane's row ----
        const v4f* rowp = (const v4f*)&s_in[p][lane * STR];  // 16B-aligned
        v4f*       orow = (v4f*)&s_out[lane * STR];
#pragma unroll
        for (int q = 0; q < TC / 4; ++q) {
            const v4f xv = rowp[q];
            v4f sv;
#define LIF_STEP(comp)                                          \
            {                                                   \
                mem = fmaf(mem, beta, xv.comp) - spk * vthc;    \
                spk = (mem - vthc) > 0.0f ? 1.0f : 0.0f;        \
                sv.comp = spk;                                  \
            }
            LIF_STEP(x) LIF_STEP(y) LIF_STEP(z) LIF_STEP(w)
#undef LIF_STEP
            orow[q] = sv;
        }

        cfence();  // order s_out ds_stores before cross-lane ds_loads below
                   // (same-wave DS ops are in-order in HW; fence is for the compiler)

        // ---- flush spikes: coalesced 2x256B contiguous segments per store ----
        float* ob = oblock + k * TC;
#pragma unroll
        for (int j = 0; j < 16; ++j) {
            const int r = 2 * j + rhalf;
            const v4f v = *(const v4f*)&s_out[r * STR + vlane * 4];
            *(v4f*)(ob + (size_t)r * T_ + vlane * 4) = v;
        }
        // next iteration's async-load asm (memory clobber) orders buffer reuse
    }
}

extern "C" void kernel_launch(void* const* d_in, const int* in_sizes, int n_in,
                              void* d_out, int out_size, void* d_ws, size_t ws_size,
                              hipStream_t stream) {
    const float* x    = (const float*)d_in[0];  // (B, C, T) f32
    const float* beta = (const float*)d_in[1];  // scalar f32
    const float* vth  = (const float*)d_in[2];  // (C,) f32
    float*       out  = (float*)d_out;          // (B, C, T) f32

    (void)in_sizes; (void)n_in; (void)out_size; (void)d_ws; (void)ws_size;

    dim3 grid((B_ * C_) / NROW);  // 1024 single-wave blocks
    dim3 block(NROW);             // 32 threads = 1 wave
    lif_scan_kernel<<<grid, block, 0, stream>>>(x, beta, vth, out);
}